// SparseMultiheadAttention_429496729817
// MI455X (gfx1250) — compile-verified
//
#include <hip/hip_runtime.h>
#include <hip/hip_bf16.h>
#include <float.h>

// Problem constants (from the reference)
#define B_  2
#define N_  10000
#define M_  320000
#define D_  256       // QD = KD = HD
#define H_  8         // NUM_HEADS

#define MAXPART_CELLS 256
#define MAXPART_STRIDE 16   // 16 uints = 64B -> one cell per cacheline

typedef float v2f __attribute__((ext_vector_type(2)));
typedef float v8f __attribute__((ext_vector_type(8)));

// ---- order-preserving float <-> uint encoding for atomic max ----
__device__ __forceinline__ unsigned f2ord(float f) {
    unsigned u = __float_as_uint(f);
    return (u & 0x80000000u) ? ~u : (u | 0x80000000u);
}
__device__ __forceinline__ float ord2f(unsigned u) {
    return (u & 0x80000000u) ? __uint_as_float(u ^ 0x80000000u)
                             : __uint_as_float(~u);
}

// ---------------------------------------------------------------------------
// Kernel 0: init segment sums + padded max-partial cells
// ---------------------------------------------------------------------------
__global__ void __launch_bounds__(256)
sma_init(float* __restrict__ seg, unsigned* __restrict__ maxpart) {
    int i = blockIdx.x * blockDim.x + threadIdx.x;
    const int total = B_ * N_ * H_;
    if (i < total) seg[i] = 0.0f;
    if (i < MAXPART_CELLS) maxpart[i * MAXPART_STRIDE] = f2ord(-FLT_MAX);
}

// ---------------------------------------------------------------------------
// Kernel 1: q = x_q @ w_q and k = x_k @ w_k via V_WMMA_F32_16X16X4_F32.
// One wave per 16x16 output tile (layouts per CDNA5 ISA §7.12.2).
// ---------------------------------------------------------------------------
__global__ void __launch_bounds__(256)
sma_gemm_qk(const float* __restrict__ xq, const float* __restrict__ xk,
            const float* __restrict__ wq, const float* __restrict__ wk,
            float* __restrict__ qout, float* __restrict__ kout) {
    const int ROWT = N_ / 16;                 // 625
    const int COLT = D_ / 16;                 // 16
    const int tilesPerMat = B_ * ROWT * COLT; // 20000

    int wid = blockIdx.x * (blockDim.x >> 5) + (threadIdx.x >> 5);
    int mat = wid / tilesPerMat;              // 0 = Q, 1 = K (wave-uniform)
    int rem = wid - mat * tilesPerMat;
    int batch = rem / (ROWT * COLT);
    int t = rem - batch * (ROWT * COLT);
    int rowTile = t / COLT;
    int colTile = t - rowTile * COLT;

    const float* X   = (mat ? xk : xq) + (size_t)batch * N_ * D_;
    const float* W   =  mat ? wk : wq;
    float*       OUT = (mat ? kout : qout) + (size_t)batch * N_ * D_;

    const int lane = threadIdx.x & 31;
    const int half = lane >> 4;               // 0: K+0/1, 1: K+2/3
    const int l15  = lane & 15;

    const float2* xrow = (const float2*)(X + (size_t)(rowTile * 16 + l15) * D_);
    const float*  wcol = W + colTile * 16 + l15;

    v8f acc = {};
#pragma unroll 4
    for (int kk = 0; kk < D_; kk += 4) {
        int ka = kk + half * 2;               // even
        float2 av = xrow[ka >> 1];
        v2f a; a.x = av.x; a.y = av.y;
        v2f b; b.x = wcol[(size_t)ka * D_];
               b.y = wcol[(size_t)(ka + 1) * D_];
        acc = __builtin_amdgcn_wmma_f32_16x16x4_f32(
                  false, a, false, b, (short)0, acc, false, false);
    }

    float* out0 = OUT + (size_t)(rowTile * 16 + half * 8) * D_ + colTile * 16 + l15;
#pragma unroll
    for (int r = 0; r < 8; ++r)
        out0[(size_t)r * D_] = acc[r];
}

// ---------------------------------------------------------------------------
// Kernel 2: per-edge scores. One wave32 per edge (b, m).
// Head sums via group-of-4 shuffles; block max via LDS; ONE atomicMax per
// block into a cacheline-padded partial cell (blockIdx & 255) -> no hot line.
// ---------------------------------------------------------------------------
__global__ void __launch_bounds__(256)
sma_edge_score(const float* __restrict__ q, const float* __restrict__ k,
               const long long* __restrict__ mask, float* __restrict__ aout,
               unsigned* __restrict__ maxpart) {
    __shared__ unsigned smax[8];

    int wid  = blockIdx.x * (blockDim.x >> 5) + (threadIdx.x >> 5); // edge id
    int lane = threadIdx.x & 31;
    int wix  = threadIdx.x >> 5;
    int b = wid / M_;
    int m = wid - b * M_;
    int i0 = (int)mask[m];        // mask[0][m]
    int i1 = (int)mask[M_ + m];   // mask[1][m]

    const float4* qr = (const float4*)(q + ((size_t)b * N_ + i0) * D_) + lane * 2;
    const float4* kr = (const float4*)(k + ((size_t)b * N_ + i1) * D_) + lane * 2;
    float4 q0 = qr[0], q1 = qr[1];
    float4 k0 = kr[0], k1 = kr[1];
    float s = q0.x * k0.x + q0.y * k0.y + q0.z * k0.z + q0.w * k0.w
            + q1.x * k1.x + q1.y * k1.y + q1.z * k1.z + q1.w * k1.w;
    // reduce within group of 4 lanes (one head = 32 contiguous dims)
    s += __shfl_xor(s, 1);
    s += __shfl_xor(s, 2);
    float a = s * (1.0f / 16.0f); // 1/sqrt(key_dim=256)
    if ((lane & 3) == 0)
        aout[(size_t)wid * H_ + (lane >> 2)] = a;

    // wave max over the 8 head leaders (lane 0 coset == leaders)
    float v = ((lane & 3) == 0) ? a : -FLT_MAX;
    v = fmaxf(v, __shfl_xor(v, 4));
    v = fmaxf(v, __shfl_xor(v, 8));
    v = fmaxf(v, __shfl_xor(v, 16));
    if (lane == 0) smax[wix] = f2ord(v);
    __syncthreads();

    // block max + single atomic into padded partial cell
    if (threadIdx.x == 0) {
        unsigned bm = smax[0];
#pragma unroll
        for (int w = 1; w < 8; ++w) bm = bm > smax[w] ? bm : smax[w];
        atomicMax(&maxpart[(blockIdx.x & (MAXPART_CELLS - 1)) * MAXPART_STRIDE], bm);
    }
}

// ---------------------------------------------------------------------------
// Kernel 2b: fold 256 padded partial maxima into the final max cell (1 block)
// ---------------------------------------------------------------------------
__global__ void __launch_bounds__(256)
sma_max_finalize(const unsigned* __restrict__ maxpart, unsigned* __restrict__ maxbuf) {
    __shared__ unsigned smax[8];
    int t = threadIdx.x;
    unsigned v = maxpart[t * MAXPART_STRIDE];
    v = max(v, (unsigned)__shfl_xor((int)v, 1));
    v = max(v, (unsigned)__shfl_xor((int)v, 2));
    v = max(v, (unsigned)__shfl_xor((int)v, 4));
    v = max(v, (unsigned)__shfl_xor((int)v, 8));
    v = max(v, (unsigned)__shfl_xor((int)v, 16));
    if ((t & 31) == 0) smax[t >> 5] = v;
    __syncthreads();
    if (t == 0) {
        unsigned bm = smax[0];
#pragma unroll
        for (int w = 1; w < 8; ++w) bm = bm > smax[w] ? bm : smax[w];
        maxbuf[0] = bm;
    }
}

// ---------------------------------------------------------------------------
// Kernel 3: e = exp(a - globalMax), scatter-add into segment sums.
// ---------------------------------------------------------------------------
__global__ void __launch_bounds__(256)
sma_exp_scatter(float* __restrict__ out, const long long* __restrict__ mask,
                float* __restrict__ seg, const unsigned* __restrict__ maxbuf) {
    int gid = blockIdx.x * blockDim.x + threadIdx.x;
    const int total = B_ * M_ * H_;           // 5,120,000
    if (gid >= total) return;
    int b = gid / (M_ * H_);
    int r = gid - b * (M_ * H_);
    int m = r / H_;
    int h = r - m * H_;
    float mx = ord2f(maxbuf[0]);
    float e = __expf(out[gid] - mx);
    out[gid] = e;
    int idx = (int)mask[m];                   // mask[0][m]
    atomicAdd(&seg[((size_t)b * N_ + idx) * H_ + h], e);
}

// ---------------------------------------------------------------------------
// Kernel 4: out = e / (seg[b, mask0[m], h] + 1e-16)
// ---------------------------------------------------------------------------
__global__ void __launch_bounds__(256)
sma_normalize(float* __restrict__ out, const long long* __restrict__ mask,
              const float* __restrict__ seg) {
    int gid = blockIdx.x * blockDim.x + threadIdx.x;
    const int total = B_ * M_ * H_;
    if (gid >= total) return;
    int b = gid / (M_ * H_);
    int r = gid - b * (M_ * H_);
    int m = r / H_;
    int h = r - m * H_;
    int idx = (int)mask[m];
    out[gid] = out[gid] / (seg[((size_t)b * N_ + idx) * H_ + h] + 1e-16f);
}

// ---------------------------------------------------------------------------
extern "C" void kernel_launch(void* const* d_in, const int* in_sizes, int n_in,
                              void* d_out, int out_size, void* d_ws, size_t ws_size,
                              hipStream_t stream) {
    const float*     xq   = (const float*)d_in[0];      // (B, N, 256)
    const float*     xk   = (const float*)d_in[1];      // (B, N, 256)
    const long long* mask = (const long long*)d_in[2];  // (2, M) int64
    const float*     wq   = (const float*)d_in[3];      // (256, 256)
    const float*     wk   = (const float*)d_in[4];      // (256, 256)
    float*           out  = (float*)d_out;              // (B, M, H)

    // workspace layout
    float* q   = (float*)d_ws;                          // B*N*D
    float* k   = q + (size_t)B_ * N_ * D_;              // B*N*D
    float* seg = k + (size_t)B_ * N_ * D_;              // B*N*H
    unsigned* maxpart = (unsigned*)(seg + (size_t)B_ * N_ * H_); // 256*16 uints
    unsigned* maxbuf  = maxpart + MAXPART_CELLS * MAXPART_STRIDE;

    // 0) init
    {
        int total = B_ * N_ * H_;
        sma_init<<<(total + 255) / 256, 256, 0, stream>>>(seg, maxpart);
    }
    // 1) q/k projections via f32 WMMA: 2 mats * 2 batches * 625 * 16 = 40000 waves
    {
        int waves = 2 * B_ * (N_ / 16) * (D_ / 16);     // 40000
        sma_gemm_qk<<<waves / 8, 256, 0, stream>>>(xq, xk, wq, wk, q, k);
    }
    // 2) edge scores: one wave per edge, B*M = 640000 waves -> 80000 blocks
    {
        int waves = B_ * M_;
        sma_edge_score<<<waves / 8, 256, 0, stream>>>(q, k, mask, out, maxpart);
    }
    // 2b) fold 256 partial maxima -> maxbuf[0]
    sma_max_finalize<<<1, 256, 0, stream>>>(maxpart, maxbuf);
    // 3) exp + segment scatter-add
    {
        int total = B_ * M_ * H_;
        sma_exp_scatter<<<(total + 255) / 256, 256, 0, stream>>>(out, mask, seg, maxbuf);
    }
    // 4) normalize
    {
        int total = B_ * M_ * H_;
        sma_normalize<<<(total + 255) / 256, 256, 0, stream>>>(out, mask, seg);
    }
}